// DomainDiscrepancyLoss_53652731461797
// MI455X (gfx1250) — compile-verified
//
#include <hip/hip_runtime.h>

typedef __attribute__((ext_vector_type(2))) float v2f;
typedef __attribute__((ext_vector_type(8))) float v8f;
typedef __attribute__((ext_vector_type(4))) int   v4i;
typedef __attribute__((address_space(1))) v4i* gv4i_p;  // global int4*
typedef __attribute__((address_space(3))) v4i* lv4i_p;  // LDS int4*

#define NROWS 4096
#define DIMS  2048
#define GAMMA 1.0f
#define BT    128   // block tile (128x128 of C per workgroup)
#define KB    32    // K-block staged in LDS
#define LDSTR 36    // padded LDS row stride (floats): 16B-aligned, bank-spread

// ---- CDNA5 async global->LDS path (ASYNCcnt-tracked), with sync fallback ----
#if defined(__has_builtin)
#if __has_builtin(__builtin_amdgcn_global_load_async_to_lds_b128)
#define USE_ASYNC_LDS 1
#endif
#endif
#ifndef USE_ASYNC_LDS
#define USE_ASYNC_LDS 0
#endif

__device__ __forceinline__ void copy16_to_lds(const float* __restrict__ g, float* l) {
#if USE_ASYNC_LDS
    // global_load_async_to_lds_b128: 16B per lane, no VGPR data path
    __builtin_amdgcn_global_load_async_to_lds_b128(
        (gv4i_p)g, (lv4i_p)l, /*imm offset*/ 0, /*cpol*/ 0);
#else
    *(float4*)l = *(const float4*)g;
#endif
}

template <int N>
__device__ __forceinline__ void wait_async_prefetch() {
#if USE_ASYNC_LDS
#if __has_builtin(__builtin_amdgcn_s_wait_asynccnt)
    __builtin_amdgcn_s_wait_asynccnt(N);
#else
    asm volatile("s_wait_asynccnt %0" ::"i"(N) : "memory");
#endif
#endif
}

// ---------------------------------------------------------------------------
// out[0] = 0
__global__ void mmd_zero_kernel(float* out) {
    if (threadIdx.x == 0 && blockIdx.x == 0) out[0] = 0.0f;
}

// ---------------------------------------------------------------------------
// norms[row] = sum_k X[row][k]^2 ; one wave (32 lanes) per row, 8 rows/block
__global__ __launch_bounds__(256) void mmd_row_norms_kernel(
    const float* __restrict__ X, float* __restrict__ norms)
{
    const int lane = threadIdx.x & 31;
    const int wave = threadIdx.x >> 5;
    const int row  = blockIdx.x * 8 + wave;

    const float4* p = (const float4*)(X + (size_t)row * DIMS);
    float s = 0.0f;
#pragma unroll 4
    for (int i = lane; i < DIMS / 4; i += 32) {
        float4 v = p[i];
        s += v.x * v.x + v.y * v.y + v.z * v.z + v.w * v.w;
    }
#pragma unroll
    for (int off = 16; off > 0; off >>= 1) s += __shfl_down(s, off, 32);
    if (lane == 0) norms[row] = s;
}

// ---------------------------------------------------------------------------
// For a 128x128 tile of C = A * B^T (A,B row-major [4096][2048]):
//   accumulate sum_{i,j in tile} exp(-g*(a2[i]+b2[j]-2*C_ij)/D^2)
// into out[0] scaled by `weight`, via V_WMMA_F32_16X16X4_F32 with
// double-buffered LDS fed by async global->LDS copies.
__global__ __launch_bounds__(256) void mmd_gemm_exp_kernel(
    const float* __restrict__ A, const float* __restrict__ B,
    const float* __restrict__ a2, const float* __restrict__ b2,
    float weight, float* __restrict__ out)
{
    __shared__ float lds_a[2][BT][LDSTR];
    __shared__ float lds_b[2][BT][LDSTR];
    __shared__ float waveSums[8];

    const int tid  = threadIdx.x;
    const int lane = tid & 31;
    const int wave = tid >> 5;
    const int hi   = lane >> 4;   // half-wave selects K-pair (ISA 16x4 f32 layout)
    const int lm   = lane & 15;   // M (A-frag) / N (B-frag) within subtile

    const int rowBase = blockIdx.y * BT;
    const int colBase = blockIdx.x * BT;
    const int rowOff  = (wave & 3) * 32;   // 8 waves tiled 4x2 -> 32x64 each
    const int colOff  = (wave >> 2) * 64;

    v8f acc[2][4];
#pragma unroll
    for (int i = 0; i < 2; ++i)
#pragma unroll
        for (int j = 0; j < 4; ++j) acc[i][j] = (v8f){0,0,0,0,0,0,0,0};

    // Stage one 128x32 A-tile + B-tile into buffer `buf` for K-offset `kb`:
    // 8 async b128 copies per thread (1024 float4 slots per matrix).
    auto prefetch = [&](int buf, int kb) {
#pragma unroll
        for (int q = 0; q < 4; ++q) {
            const int s  = tid + q * 256;
            const int r  = s >> 3;         // tile row
            const int c4 = (s & 7) * 4;    // float offset of this float4
            copy16_to_lds(A + (size_t)(rowBase + r) * DIMS + kb + c4, &lds_a[buf][r][c4]);
            copy16_to_lds(B + (size_t)(colBase + r) * DIMS + kb + c4, &lds_b[buf][r][c4]);
        }
    };

    prefetch(0, 0);

    const int NIT = DIMS / KB;  // 64 iterations, even -> buffer parity folds
#pragma unroll 2
    for (int it = 0; it < NIT; ++it) {
        const int cur = it & 1;
        if (it + 1 < NIT) {
            prefetch(cur ^ 1, (it + 1) * KB);   // overlap with this iter's WMMAs
            wait_async_prefetch<8>();           // in-order: current buffer done
        } else {
            wait_async_prefetch<0>();
        }
        __syncthreads();

#pragma unroll
        for (int ks = 0; ks < KB / 4; ++ks) {
            const int kk = ks * 4 + hi * 2;     // lanes 16-31 hold K+2,K+3
            v2f af[2], bf[4];
#pragma unroll
            for (int i = 0; i < 2; ++i) {
                af[i].x = lds_a[cur][rowOff + i * 16 + lm][kk];
                af[i].y = lds_a[cur][rowOff + i * 16 + lm][kk + 1];
            }
#pragma unroll
            for (int j = 0; j < 4; ++j) {
                bf[j].x = lds_b[cur][colOff + j * 16 + lm][kk];
                bf[j].y = lds_b[cur][colOff + j * 16 + lm][kk + 1];
            }
#pragma unroll
            for (int i = 0; i < 2; ++i)
#pragma unroll
                for (int j = 0; j < 4; ++j)
                    acc[i][j] = __builtin_amdgcn_wmma_f32_16x16x4_f32(
                        false, af[i], false, bf[j], (short)0, acc[i][j], false, false);
        }
        __syncthreads();  // protect `cur` buffer before it is re-staged
    }

    // Epilogue: exp(-c1*(a2+b2) + c2*xy) summed over the wave's 32x64 region.
    const float c1 = GAMMA / ((float)DIMS * (float)DIMS);
    const float c2 = 2.0f * c1;

    float a2v[2][8];
#pragma unroll
    for (int i = 0; i < 2; ++i)
#pragma unroll
        for (int r = 0; r < 8; ++r)
            a2v[i][r] = a2[rowBase + rowOff + i * 16 + hi * 8 + r];  // C-layout: M = r + 8*hi
    float b2v[4];
#pragma unroll
    for (int j = 0; j < 4; ++j)
        b2v[j] = b2[colBase + colOff + j * 16 + lm];                 // C-layout: N = lane%16

    float s = 0.0f;
#pragma unroll
    for (int i = 0; i < 2; ++i)
#pragma unroll
        for (int j = 0; j < 4; ++j) {
#pragma unroll
            for (int r = 0; r < 8; ++r) {
                const float base = -c1 * (a2v[i][r] + b2v[j]);
                s += __expf(fmaf(c2, acc[i][j][r], base));
            }
        }

#pragma unroll
    for (int off = 16; off > 0; off >>= 1) s += __shfl_down(s, off, 32);
    if (lane == 0) waveSums[wave] = s;
    __syncthreads();
    if (tid == 0) {
        float t = 0.0f;
#pragma unroll
        for (int w = 0; w < 8; ++w) t += waveSums[w];
        atomicAdd(out, weight * t);
    }
}

// ---------------------------------------------------------------------------
extern "C" void kernel_launch(void* const* d_in, const int* in_sizes, int n_in,
                              void* d_out, int out_size, void* d_ws, size_t ws_size,
                              hipStream_t stream) {
    const float* src = (const float*)d_in[0];
    const float* tgt = (const float*)d_in[1];
    float* out = (float*)d_out;
    float* a2  = (float*)d_ws;          // 4096 floats
    float* b2  = a2 + NROWS;            // 4096 floats

    mmd_zero_kernel<<<1, 32, 0, stream>>>(out);
    mmd_row_norms_kernel<<<NROWS / 8, 256, 0, stream>>>(src, a2);
    mmd_row_norms_kernel<<<NROWS / 8, 256, 0, stream>>>(tgt, b2);

    dim3 grid(NROWS / BT, NROWS / BT);
    const float scale = 1.0f / ((float)NROWS * (float)NROWS);
    mmd_gemm_exp_kernel<<<grid, 256, 0, stream>>>(src, src, a2, a2,  scale, out);  // xx
    mmd_gemm_exp_kernel<<<grid, 256, 0, stream>>>(tgt, tgt, b2, b2,  scale, out);  // yy
    mmd_gemm_exp_kernel<<<grid, 256, 0, stream>>>(src, tgt, a2, b2, -2.0f * scale, out); // xy
}